// PPOLoss_47528108098289
// MI455X (gfx1250) — compile-verified
//
#include <hip/hip_runtime.h>
#include <math.h>

typedef __attribute__((ext_vector_type(2))) float v2f;
typedef __attribute__((ext_vector_type(8))) float v8f;

namespace {
constexpr float  GAMMA_F = 0.99f;
constexpr float  GL_F    = (float)(0.99 * 0.95);   // gamma*lam in f64, cast like JAX scalar
constexpr float  EPS_F   = 1e-9f;
constexpr int    WPB     = 8;                      // waves per block; 1 wave == 1 row
constexpr int    P2_BLOCKS = 1024;                 // grid for stored-advantage pass 2
}

// ---- cheap wave32 lane exchange (no LDS) -----------------------------------
template <int CTRL>
__device__ __forceinline__ float upd_dpp_f32(float oldv, float x) {
  // invalid source lanes (out of 16-lane row) keep `oldv` (bound_ctrl=0)
  return __int_as_float(__builtin_amdgcn_update_dpp(
      __float_as_int(oldv), __float_as_int(x), CTRL, 0xf, 0xf, false));
}
__device__ __forceinline__ float readlane_f32(float x, int lane) {
  return __int_as_float(__builtin_amdgcn_readlane(__float_as_int(x), lane));
}
__device__ __forceinline__ float bcast0_f32(float x) {
  return __int_as_float(__builtin_amdgcn_readfirstlane(__float_as_int(x)));
}

// Wave-wide *suffix* scan of affine maps f_l(x) = D_l + C_l * x, composing
// f_l ∘ f_{l+1} ∘ ... ∘ f_31.  Steps 1,2,4,8 via DPP row_shl (identity fill),
// cross-row step via readlane(16) broadcast + predicated compose.
__device__ __forceinline__ void affine_suffix_scan(float& C, float& D, int lane) {
  float Ch, Dh;
#define SCAN_STEP(NN)                                   \
  Ch = upd_dpp_f32<0x100 | NN>(1.0f, C);                \
  Dh = upd_dpp_f32<0x100 | NN>(0.0f, D);                \
  D  = D + C * Dh;                                      \
  C  = C * Ch;
  SCAN_STEP(1) SCAN_STEP(2) SCAN_STEP(4) SCAN_STEP(8)
#undef SCAN_STEP
  const float C16 = readlane_f32(C, 16);   // composition of [16..31]
  const float D16 = readlane_f32(D, 16);
  if (lane < 16) { D = D + C * D16; C = C * C16; }
}

// ---- wave reductions -------------------------------------------------------
// f32 wave sum via V_WMMA_F32_16X16X4_F32 with all-ones B:
// A[m][0]=x(lane m), A[m][2]=x(lane m+16) => D[m][n] = x_m + x_{m+16} (all n).
// Columns are identical, so s is uniform per 16-lane half: finish w/ readlanes.
__device__ __forceinline__ float wave_sum_f32_wmma(float x) {
  v2f a; a[0] = x;    a[1] = 0.0f;
  v2f b; b[0] = 1.0f; b[1] = 1.0f;
  v8f c = {};
  c = __builtin_amdgcn_wmma_f32_16x16x4_f32(false, a, false, b, (short)0, c, false, false);
  float s = ((c[0] + c[1]) + (c[2] + c[3])) + ((c[4] + c[5]) + (c[6] + c[7]));
  return readlane_f32(s, 0) + readlane_f32(s, 16);   // uniform total
}

__device__ __forceinline__ double wave_sum_f64(double x) {
#pragma unroll
  for (int off = 16; off > 0; off >>= 1) x += __shfl_xor(x, off, 32);
  return x;
}

// ---- pass 1: GAE reverse affine scan + sum/sumsq partials ------------------
__global__ __launch_bounds__(WPB * 32) void gae_pass1(
    const float* __restrict__ rew, const float* __restrict__ val,
    const float* __restrict__ msk, float* __restrict__ adv_out,
    double* __restrict__ partials, int B, int T) {
  __shared__ double lsum[WPB], lsq[WPB];
  const int lane = threadIdx.x & 31;
  const int wid  = threadIdx.x >> 5;
  const int row  = blockIdx.x * WPB + wid;

  float fs = 0.0f, fq = 0.0f;
  if (row < B) {
    const float* r = rew + (size_t)row * T;
    const float* v = val + (size_t)row * (T + 1);
    const float* m = msk + (size_t)row * T;
    float* ao = adv_out ? adv_out + (size_t)row * T : nullptr;
    float carry = 0.0f;
    for (int k = (T >> 5) - 1; k >= 0; --k) {
      const int t = (k << 5) + lane;
      const float rv = r[t], mv = m[t], vv = v[t], vn = v[t + 1];
      if (k > 0) {  // next (lower-t) chunk -> global_prefetch_b8
        __builtin_prefetch(&r[t - 32], 0, 0);
        __builtin_prefetch(&m[t - 32], 0, 0);
        __builtin_prefetch(&v[t - 32], 0, 0);
      }
      float C = GL_F * mv;                       // adv[t] = D + C*adv[t+1]
      float D = rv + GAMMA_F * vn * mv - vv;
      affine_suffix_scan(C, D, lane);
      const float a = D + C * carry;
      carry = bcast0_f32(a);                     // adv at chunk base feeds next chunk
      if (ao) ao[t] = a;
      fs += a;
      fq += a * a;
    }
  }
  const float ws_ = wave_sum_f32_wmma(fs);       // EXEC all-1s here (convergent)
  const float wq_ = wave_sum_f32_wmma(fq);
  if (lane == 0) { lsum[wid] = (double)ws_; lsq[wid] = (double)wq_; }
  __syncthreads();
  if (threadIdx.x == 0) {
    double s = 0.0, q = 0.0;
    for (int i = 0; i < WPB; ++i) { s += lsum[i]; q += lsq[i]; }
    partials[2 * blockIdx.x]     = s;
    partials[2 * blockIdx.x + 1] = q;
  }
}

// ---- stats: mean, 1/(std+eps), value-loss term (single wave) ---------------
__global__ void stats_kernel(const double* __restrict__ p1, int np,
                             double* __restrict__ stats, double N) {
  const int l = threadIdx.x;
  double s = 0.0, q = 0.0;
  for (int i = l; i < np; i += 32) { s += p1[2 * i]; q += p1[2 * i + 1]; }
  s = wave_sum_f64(s);
  q = wave_sum_f64(q);
  if (l == 0) {
    const double mean = s / N;
    double var = (q - N * mean * mean) / (N - 1.0);   // ddof=1
    if (var < 0.0) var = 0.0;
    stats[0] = mean;
    stats[1] = 1.0 / (sqrt(var) + 1e-9);
    stats[2] = 0.5 * q / N;   // VALUE_LOSS_COEF * mean(adv_raw^2), since ret-v == adv
  }
}

// ---- pass 2 (stored adv, L2-resident): min-surrogate + entropy -------------
__global__ __launch_bounds__(256) void ppo_pass2_stored(
    const float* __restrict__ adv, const float* __restrict__ curr,
    const float* __restrict__ oldp, const double* __restrict__ stats,
    double* __restrict__ partials, long N) {
  __shared__ double lmin[8], lent[8];
  const float fm = (float)stats[0];
  const float fi = (float)stats[1];
  double dmin = 0.0;
  float  fent = 0.0f;
  const long stride = (long)gridDim.x * blockDim.x;
  for (long i = (long)blockIdx.x * blockDim.x + threadIdx.x; i < N; i += stride) {
    const float a = (adv[i] - fm) * fi;
    const float c = curr[i];
    const float ratio = c / (oldp[i] + EPS_F);
    const float cl = fminf(fmaxf(ratio, 0.8f), 1.2f);
    dmin += (double)fminf(ratio * a, cl * a);   // heavy-tailed -> f64 accumulate
    fent += c * logf(c + EPS_F);
  }
  dmin = wave_sum_f64(dmin);
  const float ew = wave_sum_f32_wmma(fent);
  const int lane = threadIdx.x & 31, wid = threadIdx.x >> 5;
  if (lane == 0) { lmin[wid] = dmin; lent[wid] = (double)ew; }
  __syncthreads();
  if (threadIdx.x == 0) {
    double s = 0.0, e = 0.0;
    for (int i = 0; i < 8; ++i) { s += lmin[i]; e += lent[i]; }
    partials[2 * blockIdx.x]     = s;
    partials[2 * blockIdx.x + 1] = e;
  }
}

// ---- pass 2 fallback: recompute GAE if ws can't hold adv -------------------
__global__ __launch_bounds__(WPB * 32) void ppo_pass2_recompute(
    const float* __restrict__ rew, const float* __restrict__ val,
    const float* __restrict__ msk, const float* __restrict__ curr,
    const float* __restrict__ oldp, const double* __restrict__ stats,
    double* __restrict__ partials, int B, int T) {
  __shared__ double lmin[WPB], lent[WPB];
  const int lane = threadIdx.x & 31;
  const int wid  = threadIdx.x >> 5;
  const int row  = blockIdx.x * WPB + wid;
  const float fm = (float)stats[0];
  const float fi = (float)stats[1];
  double dmin = 0.0;
  float  fent = 0.0f;
  if (row < B) {
    const float* r  = rew  + (size_t)row * T;
    const float* v  = val  + (size_t)row * (T + 1);
    const float* m  = msk  + (size_t)row * T;
    const float* cu = curr + (size_t)row * T;
    const float* ol = oldp + (size_t)row * T;
    float carry = 0.0f;
    for (int k = (T >> 5) - 1; k >= 0; --k) {
      const int t = (k << 5) + lane;
      const float rv = r[t], mv = m[t], vv = v[t], vn = v[t + 1];
      float C = GL_F * mv;
      float D = rv + GAMMA_F * vn * mv - vv;
      affine_suffix_scan(C, D, lane);
      const float a_raw = D + C * carry;
      carry = bcast0_f32(a_raw);
      const float a = (a_raw - fm) * fi;
      const float c = cu[t];
      const float ratio = c / (ol[t] + EPS_F);
      const float cl = fminf(fmaxf(ratio, 0.8f), 1.2f);
      dmin += (double)fminf(ratio * a, cl * a);
      fent += c * logf(c + EPS_F);
    }
  }
  dmin = wave_sum_f64(dmin);
  const float ew = wave_sum_f32_wmma(fent);
  if (lane == 0) { lmin[wid] = dmin; lent[wid] = (double)ew; }
  __syncthreads();
  if (threadIdx.x == 0) {
    double s = 0.0, e = 0.0;
    for (int i = 0; i < WPB; ++i) { s += lmin[i]; e += lent[i]; }
    partials[2 * blockIdx.x]     = s;
    partials[2 * blockIdx.x + 1] = e;
  }
}

// ---- final: 4 scalars ------------------------------------------------------
__global__ void final_kernel(const double* __restrict__ p2, int np,
                             const double* __restrict__ stats,
                             float* __restrict__ out, double N) {
  const int l = threadIdx.x;
  double nm = 0.0, en = 0.0;
  for (int i = l; i < np; i += 32) { nm += p2[2 * i]; en += p2[2 * i + 1]; }
  nm = wave_sum_f64(nm);
  en = wave_sum_f64(en);
  if (l == 0) {
    const double ppo      = -nm / N;
    const double ent_term = 0.01 * (-en / N);
    const double vterm    = stats[2];
    out[0] = (float)(ppo + vterm + ent_term);
    out[1] = (float)ppo;
    out[2] = (float)vterm;
    out[3] = (float)ent_term;
  }
}

extern "C" void kernel_launch(void* const* d_in, const int* in_sizes, int n_in,
                              void* d_out, int out_size, void* d_ws, size_t ws_size,
                              hipStream_t stream) {
  const float* rewards = (const float*)d_in[0];
  const float* values  = (const float*)d_in[1];
  // d_in[2] = ref_probs: unused by the reference
  const float* oldp    = (const float*)d_in[3];
  const float* currp   = (const float*)d_in[4];
  const float* masks   = (const float*)d_in[5];

  const int  Bn = in_sizes[1] - in_sizes[0];        // values has B extra elements
  const int  Tn = (int)((long)in_sizes[0] / Bn);
  const long N  = (long)Bn * Tn;
  const int  grid1 = (Bn + WPB - 1) / WPB;

  // workspace layout (all regions fully written before read; no atomics)
  char*   wsb   = (char*)d_ws;
  double* p1    = (double*)wsb;
  size_t  off   = (size_t)grid1 * 2 * sizeof(double);
  double* stats = (double*)(wsb + off);
  off += 4 * sizeof(double);
  const size_t p2cnt = (size_t)((P2_BLOCKS > grid1) ? P2_BLOCKS : grid1);
  double* p2 = (double*)(wsb + off);
  off += p2cnt * 2 * sizeof(double);
  off = (off + 255) & ~(size_t)255;
  float* adv = (float*)(wsb + off);
  const bool stored = (ws_size >= off + (size_t)N * sizeof(float));  // 32MB -> L2-resident

  gae_pass1<<<grid1, WPB * 32, 0, stream>>>(rewards, values, masks,
                                            stored ? adv : (float*)nullptr, p1, Bn, Tn);
  stats_kernel<<<1, 32, 0, stream>>>(p1, grid1, stats, (double)N);
  if (stored) {
    ppo_pass2_stored<<<P2_BLOCKS, 256, 0, stream>>>(adv, currp, oldp, stats, p2, N);
    final_kernel<<<1, 32, 0, stream>>>(p2, P2_BLOCKS, stats, (float*)d_out, (double)N);
  } else {
    ppo_pass2_recompute<<<grid1, WPB * 32, 0, stream>>>(rewards, values, masks, currp,
                                                        oldp, stats, p2, Bn, Tn);
    final_kernel<<<1, 32, 0, stream>>>(p2, grid1, stats, (float*)d_out, (double)N);
  }
}